// GAE_51728586113708
// MI455X (gfx1250) — compile-verified
//
#include <hip/hip_runtime.h>

typedef float v2f __attribute__((ext_vector_type(2)));
typedef float v8f __attribute__((ext_vector_type(8)));

#define CH 256  // hidden width, fixed by the model

// ---------------------------------------------------------------------------
// GEMM: out[M x 256] = A[M x K] @ W[K x 256] (+ bias) (opt relu)
// One wave per 16x16 output tile, V_WMMA_F32_16X16X4_F32 along K.
// A frag (16x4 f32): lanes 0-15 -> M=lane, {K,K+1}; lanes 16-31 -> M=lane-16, {K+2,K+3}
// B frag (4x16 f32): lanes 0-15 -> N=lane, rows {K,K+1}; lanes 16-31 -> rows {K+2,K+3}
// C/D (16x16 f32):   VGPR r: lanes 0-15 -> M=r, lanes 16-31 -> M=r+8; N = lane&15
//
// Rows >= M are handled by CLAMPING the A address to row 0 (garbage data only
// reaches D rows >= M, which the store guard discards) -> no EXEC divergence
// in the inner loop.
// ---------------------------------------------------------------------------
__global__ __launch_bounds__(256) void gemm_wmma_f32(
    const float* __restrict__ A, const float* __restrict__ W,
    float* __restrict__ out, int M, int K,
    const float* __restrict__ bias, int relu)
{
  const int lane = threadIdx.x & 31;
  const int wave = threadIdx.x >> 5;
  const long tile = (long)blockIdx.x * 8 + wave;
  const int  mtiles = (M + 15) >> 4;
  if (tile >= (long)mtiles * (CH / 16)) return;  // wave-uniform exit: EXEC stays all-1s
  const int mt = (int)(tile >> 4);
  const int nt = (int)(tile & 15);
  const int m0 = mt << 4, n0 = nt << 4;
  const int lrow = lane & 15;
  const bool hi = lane >= 16;
  const int kofs = hi ? 2 : 0;
  const int row = m0 + lrow;
  const int arow = (row < M) ? row : 0;  // clamp: avoids OOB reads, no divergence
  const int col = n0 + lrow;
  const float* ap = A + (size_t)arow * K + kofs;
  const float* bp = W + (size_t)kofs * CH + col;

  v8f c = {};
#pragma unroll 8
  for (int k = 0; k < K; k += 4) {
    v2f a, b;
    a.x = ap[0];
    a.y = ap[1];
    b.x = bp[0];
    b.y = bp[CH];
    c = __builtin_amdgcn_wmma_f32_16x16x4_f32(false, a, false, b, (short)0, c, false, false);
    ap += 4;
    bp += 4 * CH;
  }

  const float bv = bias ? bias[col] : 0.0f;
  const int rbase = m0 + (hi ? 8 : 0);
#pragma unroll
  for (int r = 0; r < 8; ++r) {
    int rr = rbase + r;
    if (rr < M) {
      float v = c[r] + bv;
      if (relu) v = fmaxf(v, 0.0f);
      out[(size_t)rr * CH + col] = v;
    }
  }
}

// ---------------------------------------------------------------------------
// Small helpers
// ---------------------------------------------------------------------------
__global__ void fill_f32(float* __restrict__ p, float v, long n) {
  long t = (long)blockIdx.x * blockDim.x + threadIdx.x;
  if (t < n) p[t] = v;
}

__global__ void deg_accum(const int* __restrict__ dst, int E, float* __restrict__ deg) {
  int e = blockIdx.x * blockDim.x + threadIdx.x;
  if (e < E) atomicAdd(&deg[dst[e]], 1.0f);
}

__global__ void rsqrt_k(const float* __restrict__ deg, float* __restrict__ inv, int n) {
  int i = blockIdx.x * blockDim.x + threadIdx.x;
  if (i < n) inv[i] = rsqrtf(fmaxf(deg[i], 1e-12f));
}

// acc[i,:] = h[i,:] * inv[i]^2   (self-loop term), float4 per thread
__global__ void self_init(const float* __restrict__ h, const float* __restrict__ inv,
                          float* __restrict__ acc, int n) {
  long t = (long)blockIdx.x * blockDim.x + threadIdx.x;
  if (t >= (long)n * (CH / 4)) return;
  int i = (int)(t >> 6);
  float w = inv[i];
  w *= w;
  float4 hv = ((const float4*)h)[t];
  float4 o;
  o.x = hv.x * w; o.y = hv.y * w; o.z = hv.z * w; o.w = hv.w * w;
  ((float4*)acc)[t] = o;
}

// acc[dst,:] += h[src,:] * inv[src]*inv[dst]   via f32 atomics, 4 channels/thread
__global__ void scatter_edges(const int* __restrict__ src, const int* __restrict__ dst, int E,
                              const float* __restrict__ h, const float* __restrict__ inv,
                              float* __restrict__ acc) {
  long t = (long)blockIdx.x * blockDim.x + threadIdx.x;
  if (t >= (long)E * (CH / 4)) return;
  int e = (int)(t >> 6);
  int g = ((int)t & 63) << 2;
  int s = src[e], d = dst[e];
  float w = inv[s] * inv[d];
  const float* hp = h + (size_t)s * CH + g;
  float* ap = acc + (size_t)d * CH + g;
  atomicAdd(ap + 0, hp[0] * w);
  atomicAdd(ap + 1, hp[1] * w);
  atomicAdd(ap + 2, hp[2] * w);
  atomicAdd(ap + 3, hp[3] * w);
}

__global__ void bias_act(const float* __restrict__ acc, const float* __restrict__ b,
                         float* __restrict__ out, int n, int relu) {
  long t = (long)blockIdx.x * blockDim.x + threadIdx.x;
  if (t >= (long)n * (CH / 4)) return;
  int g = (int)t & 63;
  float4 v = ((const float4*)acc)[t];
  float4 bb = ((const float4*)b)[g];
  v.x += bb.x; v.y += bb.y; v.z += bb.z; v.w += bb.w;
  if (relu) {
    v.x = fmaxf(v.x, 0.0f); v.y = fmaxf(v.y, 0.0f);
    v.z = fmaxf(v.z, 0.0f); v.w = fmaxf(v.w, 0.0f);
  }
  ((float4*)out)[t] = v;
}

// out[r,:] = in[perm[r],:]   (256-wide rows, float4)
__global__ void gather_rows(const float* __restrict__ in, const int* __restrict__ perm,
                            float* __restrict__ out, int nrows) {
  long t = (long)blockIdx.x * blockDim.x + threadIdx.x;
  if (t >= (long)nrows * (CH / 4)) return;
  int r = (int)(t >> 6), g = (int)t & 63;
  ((float4*)out)[t] = ((const float4*)in)[(long)perm[r] * (CH / 4) + g];
}

// up[perm[r],:] = x[r,:]   (up must be pre-zeroed)
__global__ void scatter_rows(const float* __restrict__ x, const int* __restrict__ perm,
                             float* __restrict__ up, int nrows) {
  long t = (long)blockIdx.x * blockDim.x + threadIdx.x;
  if (t >= (long)nrows * (CH / 4)) return;
  int r = (int)(t >> 6), g = (int)t & 63;
  ((float4*)up)[(long)perm[r] * (CH / 4) + g] = ((const float4*)x)[t];
}

__global__ void gather_pos(const float* __restrict__ pos, const int* __restrict__ perm,
                           float* __restrict__ out, int nrows) {
  int r = blockIdx.x * blockDim.x + threadIdx.x;
  if (r < nrows) {
    int p = perm[r];
    out[2 * r + 0] = pos[2 * p + 0];
    out[2 * r + 1] = pos[2 * p + 1];
  }
}

// ---------------------------------------------------------------------------
// Brute-force KNN: one block per query row; k rounds of min-reduction with
// exclusion of already-selected neighbors. pos is [n x 2].
// ---------------------------------------------------------------------------
__global__ __launch_bounds__(256) void knn_kernel(const float* __restrict__ pos, int n, int k,
                                                  int* __restrict__ src, int* __restrict__ dst) {
  const int i = blockIdx.x;
  const int tid = threadIdx.x;
  const float xi = pos[2 * i], yi = pos[2 * i + 1];
  __shared__ float sd[256];
  __shared__ int   si[256];
  __shared__ int   sel[8];
  for (int kk = 0; kk < k; ++kk) {
    float best = 1e30f;
    int bidx = 0x7fffffff;
    for (int j = tid; j < n; j += 256) {
      if (j == i) continue;
      bool used = false;
      for (int t = 0; t < kk; ++t) used |= (sel[t] == j);
      if (used) continue;
      float dx = xi - pos[2 * j];
      float dy = yi - pos[2 * j + 1];
      float d = dx * dx + dy * dy;
      if (d < best || (d == best && j < bidx)) { best = d; bidx = j; }
    }
    sd[tid] = best;
    si[tid] = bidx;
    __syncthreads();
    for (int s = 128; s > 0; s >>= 1) {
      if (tid < s) {
        float dv = sd[tid + s];
        int iv = si[tid + s];
        if (dv < sd[tid] || (dv == sd[tid] && iv < si[tid])) { sd[tid] = dv; si[tid] = iv; }
      }
      __syncthreads();
    }
    if (tid == 0) {
      int b = si[0];
      sel[kk] = b;
      src[(long)i * k + kk] = b;
      dst[(long)i * k + kk] = i;
    }
    __syncthreads();
  }
}

// ---------------------------------------------------------------------------
// Host orchestration
// ---------------------------------------------------------------------------
extern "C" void kernel_launch(void* const* d_in, const int* in_sizes, int n_in,
                              void* d_out, int out_size, void* d_ws, size_t ws_size,
                              hipStream_t stream) {
  const float* x    = (const float*)d_in[0];
  const float* pos  = (const float*)d_in[1];
  const int*   eidx = (const int*)d_in[2];
  const int* perm1 = (const int*)d_in[3];
  const int* perm2 = (const int*)d_in[4];
  const int* perm3 = (const int*)d_in[5];
  const float* Wd0 = (const float*)d_in[6],  *bd0 = (const float*)d_in[7];
  const float* Wd1 = (const float*)d_in[8],  *bd1 = (const float*)d_in[9];
  const float* Wd2 = (const float*)d_in[10], *bd2 = (const float*)d_in[11];
  const float* Wu0 = (const float*)d_in[12], *bu0 = (const float*)d_in[13];
  const float* Wu1 = (const float*)d_in[14], *bu1 = (const float*)d_in[15];
  const float* Wu2 = (const float*)d_in[16], *bu2 = (const float*)d_in[17];
  const float* Wlin = (const float*)d_in[18], *blin = (const float*)d_in[19];

  const int N0 = 40000, N1 = 10000, N2 = 2500, N3 = 625;
  const int CIN = 64;
  const int E0 = in_sizes[2] / 2;
  const int K1 = 6, K2 = 7;
  const int E1 = N1 * K1, E2 = N2 * K2;

  char* wp = (char*)d_ws;
  auto alloc = [&](size_t bytes) -> void* {
    void* r = (void*)wp;
    wp += (bytes + 255) & ~(size_t)255;
    return r;
  };
  float* BA   = (float*)alloc((size_t)N0 * CH * sizeof(float));
  float* BB   = (float*)alloc((size_t)N0 * CH * sizeof(float));
  float* deg  = (float*)alloc((size_t)N0 * sizeof(float));
  float* inv  = (float*)alloc((size_t)N0 * sizeof(float));
  float* pos1 = (float*)alloc((size_t)N1 * 2 * sizeof(float));
  float* pos2 = (float*)alloc((size_t)N2 * 2 * sizeof(float));
  int* src1 = (int*)alloc((size_t)E1 * sizeof(int));
  int* dst1 = (int*)alloc((size_t)E1 * sizeof(int));
  int* src2 = (int*)alloc((size_t)E2 * sizeof(int));
  int* dst2 = (int*)alloc((size_t)E2 * sizeof(int));
  float* D = (float*)d_out;  // use d_out as a third large buffer

  const int* esrc0 = eidx;
  const int* edst0 = eidx + E0;

  auto conv = [&](const float* xin, int n, int K, const float* W, const float* b,
                  const int* es, const int* ed, int E, float* h, float* acc, float* o) {
    long tiles = (long)((n + 15) / 16) * (CH / 16);
    gemm_wmma_f32<<<(unsigned)((tiles + 7) / 8), 256, 0, stream>>>(xin, W, h, n, K, nullptr, 0);
    fill_f32<<<(n + 255) / 256, 256, 0, stream>>>(deg, 1.0f, n);
    deg_accum<<<(E + 255) / 256, 256, 0, stream>>>(ed, E, deg);
    rsqrt_k<<<(n + 255) / 256, 256, 0, stream>>>(deg, inv, n);
    long nt = (long)n * (CH / 4);
    self_init<<<(unsigned)((nt + 255) / 256), 256, 0, stream>>>(h, inv, acc, n);
    long et = (long)E * (CH / 4);
    scatter_edges<<<(unsigned)((et + 255) / 256), 256, 0, stream>>>(es, ed, E, h, inv, acc);
    bias_act<<<(unsigned)((nt + 255) / 256), 256, 0, stream>>>(acc, b, o, n, 1);
  };

  // ---- encoder level 0 (N0, original edges) ----
  conv(x, N0, CIN, Wd0, bd0, esrc0, edst0, E0, BA, BB, D);
  // pool 1 + KNN rebuild at N1 (k=6)
  gather_rows<<<(unsigned)(((long)N1 * 64 + 255) / 256), 256, 0, stream>>>(D, perm1, BA, N1);
  gather_pos<<<(N1 + 255) / 256, 256, 0, stream>>>(pos, perm1, pos1, N1);
  knn_kernel<<<N1, 256, 0, stream>>>(pos1, N1, K1, src1, dst1);
  // ---- encoder level 1 ----
  conv(BA, N1, CH, Wd1, bd1, src1, dst1, E1, BB, D, BA);
  // pool 2 + KNN rebuild at N2 (k=7)
  gather_rows<<<(unsigned)(((long)N2 * 64 + 255) / 256), 256, 0, stream>>>(BA, perm2, BB, N2);
  gather_pos<<<(N2 + 255) / 256, 256, 0, stream>>>(pos1, perm2, pos2, N2);
  knn_kernel<<<N2, 256, 0, stream>>>(pos2, N2, K2, src2, dst2);
  // ---- encoder level 2 ----
  conv(BB, N2, CH, Wd2, bd2, src2, dst2, E2, BA, D, BB);
  // pool 3 (x3 in BA); level-3 KNN result is dead in the reference -> skipped
  gather_rows<<<(unsigned)(((long)N3 * 64 + 255) / 256), 256, 0, stream>>>(BB, perm3, BA, N3);

  // ---- decoder i=0 at N2, edges=knn2, Wu0 ----
  fill_f32<<<(unsigned)(((long)N2 * CH + 255) / 256), 256, 0, stream>>>(BB, 0.0f, (long)N2 * CH);
  scatter_rows<<<(unsigned)(((long)N3 * 64 + 255) / 256), 256, 0, stream>>>(BA, perm3, BB, N3);
  conv(BB, N2, CH, Wu0, bu0, src2, dst2, E2, D, BA, BB);
  // ---- decoder i=1 at N1, edges=knn1, Wu1 ----
  fill_f32<<<(unsigned)(((long)N1 * CH + 255) / 256), 256, 0, stream>>>(D, 0.0f, (long)N1 * CH);
  scatter_rows<<<(unsigned)(((long)N2 * 64 + 255) / 256), 256, 0, stream>>>(BB, perm2, D, N2);
  conv(D, N1, CH, Wu1, bu1, src1, dst1, E1, BA, BB, D);
  // ---- decoder i=2 at N0, original edges, Wu2 ----
  fill_f32<<<(unsigned)(((long)N0 * CH + 255) / 256), 256, 0, stream>>>(BA, 0.0f, (long)N0 * CH);
  scatter_rows<<<(unsigned)(((long)N1 * 64 + 255) / 256), 256, 0, stream>>>(D, perm1, BA, N1);
  conv(BA, N0, CH, Wu2, bu2, esrc0, edst0, E0, BB, D, BA);

  // ---- final linear: out = x @ Wlin + blin ----
  long tiles = (long)((N0 + 15) / 16) * (CH / 16);
  gemm_wmma_f32<<<(unsigned)((tiles + 7) / 8), 256, 0, stream>>>(BA, Wlin, D, N0, CH, blin, 0);

  (void)n_in; (void)out_size; (void)ws_size;
}